// GCNRegression_69758858822059
// MI455X (gfx1250) — compile-verified
//
#include <hip/hip_runtime.h>

typedef float v2f __attribute__((ext_vector_type(2)));
typedef float v8f __attribute__((ext_vector_type(8)));

// ---------------- utility ----------------
__global__ __launch_bounds__(256) void k_zero(float* __restrict__ p, long long n) {
  long long i = (long long)blockIdx.x * blockDim.x + threadIdx.x;
  long long stride = (long long)gridDim.x * blockDim.x;
  for (; i < n; i += stride) p[i] = 0.0f;
}

// deg[dst] += 1 per edge
__global__ __launch_bounds__(256) void k_degree(const int* __restrict__ ei,
                                                float* __restrict__ deg, int E) {
  int e = blockIdx.x * 256 + threadIdx.x;
  if (e < E) atomicAdd(&deg[ei[E + e]], 1.0f);
}

// deg -> rsqrt(deg + 1)  (self loop)
__global__ __launch_bounds__(256) void k_dinv(float* __restrict__ deg, int N) {
  int i = blockIdx.x * 256 + threadIdx.x;
  if (i < N) deg[i] = rsqrtf(deg[i] + 1.0f);
}

// ---------------- Y[M,128] = X[M,128] @ W[128,128] via V_WMMA_F32_16X16X4_F32 ----------------
__global__ __launch_bounds__(256) void k_gemm128(const float* __restrict__ X,
                                                 const float* __restrict__ W,
                                                 float* __restrict__ Y, int M) {
  __shared__ float Ws[128 * 128];  // 64 KB weight tile in LDS
  const int tid = threadIdx.x;
  const int rowBase = blockIdx.x * 128;

  for (int i = tid * 4; i < 128 * 128; i += 256 * 4)
    *(float4*)&Ws[i] = *(const float4*)&W[i];
  __syncthreads();

  const int wave  = tid >> 5;        // 0..7 : 16-row stripe per wave
  const int lane  = tid & 31;
  const int l15   = lane & 15;
  const int khalf = (lane >> 4) << 1;  // lanes 16-31 hold K=+2,+3
  const int rowOff = (lane >> 4) << 3; // C/D: lanes 16-31 hold M=g+8

  int arow = rowBase + wave * 16 + l15;
  if (arow >= M) arow = M - 1;         // clamp: keep EXEC all-ones for WMMA
  const float* __restrict__ xrow = X + (size_t)arow * 128;

  for (int t = 0; t < 8; ++t) {        // 8 column tiles of 16
    const int colBase = t * 16;
    v8f acc = {0.f, 0.f, 0.f, 0.f, 0.f, 0.f, 0.f, 0.f};
#pragma unroll
    for (int k0 = 0; k0 < 128; k0 += 4) {
      v2f a, b;
      a.x = xrow[k0 + khalf];
      a.y = xrow[k0 + khalf + 1];
      b.x = Ws[(k0 + khalf) * 128 + colBase + l15];
      b.y = Ws[(k0 + khalf + 1) * 128 + colBase + l15];
      acc = __builtin_amdgcn_wmma_f32_16x16x4_f32(false, a, false, b,
                                                  (short)0, acc, false, false);
    }
#pragma unroll
    for (int g = 0; g < 8; ++g) {
      const int r = rowBase + wave * 16 + g + rowOff;
      if (r < M) Y[(size_t)r * 128 + colBase + l15] = acc[g];
    }
  }
}

// ---------------- edge scatter-gather: agg[dst] += h[src] * dinv[src]*dinv[dst] ----------------
// one wave per edge, float4 per lane (32 lanes * 4 = 128 features)
__global__ __launch_bounds__(256) void k_edge_agg(const int* __restrict__ ei,
                                                  const float* __restrict__ h,
                                                  const float* __restrict__ dinv,
                                                  float* __restrict__ agg, int E) {
  long long gid = (long long)blockIdx.x * 256 + threadIdx.x;
  long long e = gid >> 5;
  if (e >= E) return;
  const int lane = (int)(gid & 31);
  const int s = ei[e];
  const int d = ei[(long long)E + e];
  const float nrm = dinv[s] * dinv[d];
  const float4 hv = *(const float4*)&h[(size_t)s * 128 + lane * 4];
  float* ap = &agg[(size_t)d * 128 + lane * 4];
  atomicAdd(ap + 0, hv.x * nrm);
  atomicAdd(ap + 1, hv.y * nrm);
  atomicAdd(ap + 2, hv.z * nrm);
  atomicAdd(ap + 3, hv.w * nrm);
}

// out = relu(agg + h*dinv^2 + bias)
__global__ __launch_bounds__(256) void k_combine_relu(const float* __restrict__ agg,
                                                      const float* __restrict__ h,
                                                      const float* __restrict__ dinv,
                                                      const float* __restrict__ bias,
                                                      float* __restrict__ out,
                                                      long long total) {
  long long idx = (long long)blockIdx.x * 256 + threadIdx.x;
  if (idx >= total) return;
  const int f = (int)(idx & 127);
  const long long i = idx >> 7;
  const float di = dinv[i];
  const float v = agg[idx] + h[idx] * (di * di) + bias[f];
  out[idx] = fmaxf(v, 0.0f);
}

// ---------------- pooling ----------------
__global__ __launch_bounds__(256) void k_pool_cnt(const int* __restrict__ batch,
                                                  float* __restrict__ cnt, int N) {
  int i = blockIdx.x * 256 + threadIdx.x;
  if (i < N) atomicAdd(&cnt[batch[i]], 1.0f);
}

__global__ __launch_bounds__(256) void k_pool_sum(const int* __restrict__ batch,
                                                  const float* __restrict__ h,
                                                  float* __restrict__ sums,
                                                  long long total) {
  long long idx = (long long)blockIdx.x * 256 + threadIdx.x;
  if (idx >= total) return;
  const long long i = idx >> 7;
  const int f = (int)(idx & 127);
  atomicAdd(&sums[(size_t)batch[i] * 128 + f], h[idx]);
}

// out[g, j] = mean_pool(g) . Wlin[:, j] + blin[j]
__global__ __launch_bounds__(128) void k_final(const float* __restrict__ sums,
                                               const float* __restrict__ cnt,
                                               const float* __restrict__ Wlin,
                                               const float* __restrict__ blin,
                                               float* __restrict__ out, int G) {
  int idx = blockIdx.x * 128 + threadIdx.x;
  if (idx >= G * 2) return;
  const int g = idx >> 1, j = idx & 1;
  const float inv = 1.0f / fmaxf(cnt[g], 1.0f);
  float s = 0.0f;
  for (int f = 0; f < 128; ++f) s += sums[g * 128 + f] * Wlin[f * 2 + j];
  out[idx] = s * inv + blin[j];
}

// ---------------- host orchestration ----------------
extern "C" void kernel_launch(void* const* d_in, const int* in_sizes, int n_in,
                              void* d_out, int out_size, void* d_ws, size_t ws_size,
                              hipStream_t stream) {
  (void)n_in; (void)ws_size;
  const float* x    = (const float*)d_in[0];
  const int*   ei   = (const int*)d_in[1];    // [2, E]
  const int*   bat  = (const int*)d_in[2];    // [N]
  // d_in[3] = num_graphs scalar (derived from out_size instead)
  const float* W1   = (const float*)d_in[4];
  const float* b1   = (const float*)d_in[5];
  const float* W2   = (const float*)d_in[6];
  const float* b2   = (const float*)d_in[7];
  const float* Wlin = (const float*)d_in[8];
  const float* blin = (const float*)d_in[9];
  float* out = (float*)d_out;

  const int N = in_sizes[0] / 128;
  const int E = in_sizes[1] / 2;
  const int G = out_size / 2;
  const long long NF = (long long)N * 128;

  float* deg  = (float*)d_ws;                  // N    (becomes dinv)
  float* bufA = deg + N;                       // N*128
  float* bufB = bufA + NF;                     // N*128
  float* psum = bufB + NF;                     // G*128
  // pcnt = psum + G*128 (zeroed together with psum)
  float* pcnt = psum + (long long)G * 128;     // G

  auto cdiv = [](long long a, long long b) { return (unsigned)((a + b - 1) / b); };

  // zero accumulators (deterministic per call)
  k_zero<<<1024, 256, 0, stream>>>(deg, N);
  k_zero<<<4096, 256, 0, stream>>>(bufB, NF);
  k_zero<<<256, 256, 0, stream>>>(psum, (long long)G * 128 + G);

  // normalization
  k_degree<<<cdiv(E, 256), 256, 0, stream>>>(ei, deg, E);
  k_dinv<<<cdiv(N, 256), 256, 0, stream>>>(deg, N);

  // layer 1: h = x@W1 ; aggregate ; relu(agg + self + b1) -> bufA
  k_gemm128<<<cdiv(N, 128), 256, 0, stream>>>(x, W1, bufA, N);
  k_edge_agg<<<cdiv((long long)E * 32, 256), 256, 0, stream>>>(ei, bufA, deg, bufB, E);
  k_combine_relu<<<cdiv(NF, 256), 256, 0, stream>>>(bufB, bufA, deg, b1, bufA, NF);

  // layer 2: h = bufA@W2 -> bufB ; aggregate into (re-zeroed) bufA ; fuse -> bufB
  k_gemm128<<<cdiv(N, 128), 256, 0, stream>>>(bufA, W2, bufB, N);
  k_zero<<<4096, 256, 0, stream>>>(bufA, NF);
  k_edge_agg<<<cdiv((long long)E * 32, 256), 256, 0, stream>>>(ei, bufB, deg, bufA, E);
  k_combine_relu<<<cdiv(NF, 256), 256, 0, stream>>>(bufA, bufB, deg, b2, bufB, NF);

  // global mean pool + linear head
  k_pool_cnt<<<cdiv(N, 256), 256, 0, stream>>>(bat, pcnt, N);
  k_pool_sum<<<cdiv(NF, 256), 256, 0, stream>>>(bat, bufB, psum, NF);
  k_final<<<cdiv((long long)G * 2, 128), 128, 0, stream>>>(psum, pcnt, Wlin, blin, out, G);
}